// RelativePostionalEncoding_15375982919765
// MI455X (gfx1250) — compile-verified
//
#include <hip/hip_runtime.h>

typedef __attribute__((ext_vector_type(2))) float v2f;
typedef __attribute__((ext_vector_type(8))) float v8f;

#define B_  2
#define S_  512
#define H_  8
#define D_  64
#define E_  512
#define R_  256   // table rows actually reachable: idx = 128 + xj - xi ∈ [1,255]
#define TM  32    // query rows per block

// One block per (b, h, 32-row tile of i).
// Phase 1: P[m, r] = sum_d q[b, i0+m, h, d] * T[r, h*64+d]  via v_wmma_f32_16x16x4_f32
// Phase 2: out[b,h,i0+m,j] = P[m, mx + x[b,j] - x[b,i0+m]]
__global__ __launch_bounds__(256) void relpos_wmma_kernel(
    const float* __restrict__ q,
    const float* __restrict__ table,
    const int*   __restrict__ x,
    const int*   __restrict__ max_x_p,
    float* __restrict__ out)
{
    __shared__ float ldsQ[TM * D_];   // 8 KB : q tile, row-major [m][d]
    __shared__ float ldsP[TM * R_];   // 32 KB: P tile, row-major [m][r]
    __shared__ int   xs[S_];          // 2 KB : x[b, :]

    const int tid   = threadIdx.x;
    const int bid   = blockIdx.x;
    const int itile = bid & ((S_ / TM) - 1);       // 16 tiles
    const int h     = (bid >> 4) & (H_ - 1);
    const int b     = bid >> 7;
    const int i0    = itile * TM;
    const int mx    = max_x_p[0];

    // ---- stage q tile: 32 rows x 64 cols ----
#pragma unroll
    for (int k = 0; k < (TM * D_) / 256; ++k) {
        int idx = tid + k * 256;
        int m = idx >> 6;       // /64
        int d = idx & (D_ - 1);
        ldsQ[idx] = q[(((size_t)b * S_ + (i0 + m)) * H_ + h) * D_ + d];
    }
    // ---- stage x row ----
#pragma unroll
    for (int k = 0; k < S_ / 256; ++k) {
        int j = tid + k * 256;
        xs[j] = x[b * S_ + j];
    }
    __syncthreads();

    const int wave   = tid >> 5;          // 0..7
    const int lane   = tid & 31;
    const int laneLo = lane & 15;
    const int hiSel  = lane >> 4;         // 0: lanes 0-15 (K=0,1), 1: lanes 16-31 (K=2,3)

    // ---- preload A fragments: a[mt][k4] covers rows mt*16.., K = 4*k4..4*k4+3 ----
    // ISA f32 A 16x4 layout: lane<16 -> (K=0 in v0, K=1 in v1); lane>=16 -> (K=2, K=3)
    v2f a[2][16];
#pragma unroll
    for (int mt = 0; mt < 2; ++mt) {
        const float* src = &ldsQ[(mt * 16 + laneLo) * D_ + 2 * hiSel];
#pragma unroll
        for (int k4 = 0; k4 < 16; ++k4) {
            a[mt][k4] = *reinterpret_cast<const v2f*>(src + 4 * k4);
        }
    }

    // ---- GEMM: each wave owns 2 N-tiles (r-tiles) ----
#pragma unroll
    for (int p = 0; p < 2; ++p) {
        const int nt = wave * 2 + p;
        const int r  = nt * 16 + laneLo;
        // B 4x16 (KxN): lane holds column N=r; v0/v1 = (K=0,1) for lanes 0-15, (K=2,3) for 16-31.
        // T[r, h*64 + K]: the two K values per lane are contiguous in memory -> one b64 load.
        const float* bsrc = table + (size_t)r * E_ + h * D_ + 2 * hiSel;

        v8f acc0 = {};   // m-tile 0 (rows i0..i0+15)
        v8f acc1 = {};   // m-tile 1 (rows i0+16..i0+31)
#pragma unroll
        for (int k4 = 0; k4 < 16; ++k4) {
            v2f bv = *reinterpret_cast<const v2f*>(bsrc + 4 * k4);
            acc0 = __builtin_amdgcn_wmma_f32_16x16x4_f32(
                       false, a[0][k4], false, bv, (short)0, acc0, false, false);
            acc1 = __builtin_amdgcn_wmma_f32_16x16x4_f32(
                       false, a[1][k4], false, bv, (short)0, acc1, false, false);
        }

        // C/D layout: VGPR c holds (M = c + 8*hiSel, N = laneLo)
#pragma unroll
        for (int c = 0; c < 8; ++c) {
            int M = c + 8 * hiSel;
            ldsP[M * R_ + nt * 16 + laneLo]         = acc0[c];
            ldsP[(16 + M) * R_ + nt * 16 + laneLo]  = acc1[c];
        }
    }
    __syncthreads();

    // ---- gather + coalesced store: out[b,h,i0+m,j] ----
    float* outBase = out + (((size_t)(b * H_ + h)) * S_ + i0) * S_;
#pragma unroll 4
    for (int idx = tid; idx < TM * S_; idx += 256) {
        int m   = idx >> 9;            // /512
        int j   = idx & (S_ - 1);
        int rel = mx + xs[j] - xs[i0 + m];   // in [1,255]
        outBase[(size_t)m * S_ + j] = ldsP[m * R_ + rel];
    }
}

extern "C" void kernel_launch(void* const* d_in, const int* in_sizes, int n_in,
                              void* d_out, int out_size, void* d_ws, size_t ws_size,
                              hipStream_t stream) {
    const float* q     = (const float*)d_in[0];
    const float* table = (const float*)d_in[1];
    const int*   x     = (const int*)d_in[2];
    const int*   mx    = (const int*)d_in[3];
    float*       out   = (float*)d_out;
    (void)in_sizes; (void)n_in; (void)out_size; (void)d_ws; (void)ws_size;

    dim3 grid(B_ * H_ * (S_ / TM));   // 256 blocks
    dim3 block(256);                  // 8 waves (wave32)
    hipLaunchKernelGGL(relpos_wmma_kernel, grid, block, 0, stream, q, table, x, mx, out);
}